// Network_43808666419473
// MI455X (gfx1250) — compile-verified
//
#include <hip/hip_runtime.h>
#include <hip/hip_bf16.h>

// Problem constants (match reference)
#define BB 8
#define NN 256
#define DD 300   // = 75 * 4, exact multiple of WMMA f32 K-step
#define PP 128
#define ROWS (BB * NN)   // 2048

typedef __attribute__((ext_vector_type(2))) float v2f;
typedef __attribute__((ext_vector_type(8))) float v8f;
typedef __attribute__((ext_vector_type(4))) unsigned int u32x4;
typedef __attribute__((ext_vector_type(8))) int i32x8;
typedef __attribute__((ext_vector_type(4))) int i32x4;

// ---------------------------------------------------------------------------
// Kernel 1: embedding gather  x[row, d] = emb[sentence[row], d]
// ---------------------------------------------------------------------------
__global__ void __launch_bounds__(256)
gather_embed_kernel(const int* __restrict__ sent,
                    const float* __restrict__ emb,
                    float* __restrict__ x) {
  int e = blockIdx.x * blockDim.x + threadIdx.x;
  const int total = ROWS * DD;
  if (e >= total) return;
  int row = e / DD;
  int d = e - row * DD;
  x[e] = emb[(long)sent[row] * DD + d];
}

// ---------------------------------------------------------------------------
// Kernel 2: fused GEMM  [2048 x 300] x [300 x 512] via V_WMMA_F32_16X16X4_F32
// N-dim is the concatenation Wl | Wr | Ws1 | We1 (each 128 cols).
// One wave (32 threads) computes one 16x16 C tile; 75 K-steps of 4.
// Epilogue: bias + relu per branch, scatter to left/right/hs1/he1.
// ---------------------------------------------------------------------------
__global__ void __launch_bounds__(32)
fused_gemm_wmma(const float* __restrict__ x,
                const float* __restrict__ Wl, const float* __restrict__ Wr,
                const float* __restrict__ Ws1, const float* __restrict__ We1,
                const float* __restrict__ bl, const float* __restrict__ bs1,
                const float* __restrict__ be1,
                float* __restrict__ left, float* __restrict__ right,
                float* __restrict__ hs1, float* __restrict__ he1) {
  const int lane = threadIdx.x;      // 0..31 (wave32)
  const int half = lane >> 4;        // 0: K={0,1}, 1: K={2,3}
  const int l16  = lane & 15;
  const int ntile = blockIdx.x;      // 0..31  (32 * 16 = 512 output cols)
  const int m0    = blockIdx.y << 4; // 0..2032
  const int wsel  = ntile >> 3;      // which weight matrix (8 tiles each)
  const int ncol  = ((ntile & 7) << 4) + l16;  // 0..127 within that weight

  const float* W = (wsel == 0) ? Wl : (wsel == 1) ? Wr : (wsel == 2) ? Ws1 : We1;

  // ISA 32-bit A 16x4 layout: lanes 0-15 rows M, VGPR0/1 = K 0/1; lanes 16-31 = K 2/3.
  // B is 4x16 = Wt, i.e. B[k][n] = W[n][k]; per-lane the two K values are contiguous.
  const float* arow = x + (long)(m0 + l16) * DD + half * 2;
  const float* brow = W + (long)ncol * DD + half * 2;

  v8f c = {};
  for (int k = 0; k < DD; k += 4) {
    v2f a, b;
    a[0] = arow[k];     a[1] = arow[k + 1];
    b[0] = brow[k];     b[1] = brow[k + 1];
    // 8-arg form: (neg_a, A, neg_b, B, c_mod, C, reuse_a, reuse_b)
    c = __builtin_amdgcn_wmma_f32_16x16x4_f32(false, a, false, b,
                                              (short)0, c, false, false);
  }

  // C/D layout: VGPR v -> M = m0 + v + 8*half, N = ncol
#pragma unroll
  for (int v = 0; v < 8; ++v) {
    int m = m0 + v + (half << 3);
    long o = (long)m * PP + ncol;
    float val = c[v];
    if (wsel == 0)      left[o]  = val + bl[ncol];
    else if (wsel == 1) right[o] = val;
    else if (wsel == 2) hs1[o]   = fmaxf(val + bs1[ncol], 0.0f);
    else                he1[o]   = fmaxf(val + be1[ncol], 0.0f);
  }
}

// ---------------------------------------------------------------------------
// Kernel 3: start/end heads — wave-per-row 128-dot with shfl_xor reduction
// ---------------------------------------------------------------------------
__global__ void __launch_bounds__(256)
startend_kernel(const float* __restrict__ hs1, const float* __restrict__ he1,
                const float* __restrict__ Ws2, const float* __restrict__ We2,
                const float* __restrict__ bs2, const float* __restrict__ be2,
                float* __restrict__ start_o, float* __restrict__ end_o) {
  int wave = threadIdx.x >> 5;
  int lane = threadIdx.x & 31;
  int m = blockIdx.x * 8 + wave;   // 256 blocks * 8 waves = 2048 rows
  float s = 0.0f, e = 0.0f;
#pragma unroll
  for (int q = 0; q < 4; ++q) {
    int p = lane + (q << 5);
    s = fmaf(hs1[(long)m * PP + p], Ws2[p], s);
    e = fmaf(he1[(long)m * PP + p], We2[p], e);
  }
#pragma unroll
  for (int off = 16; off > 0; off >>= 1) {
    s += __shfl_xor(s, off, 32);
    e += __shfl_xor(e, off, 32);
  }
  if (lane == 0) {
    start_o[m] = s + bs2[0];
    end_o[m]   = e + be2[0];
  }
}

// ---------------------------------------------------------------------------
// TDM helper: DMA a 2D f32 tile [rows x 128] global -> LDS, with hardware
// padding of 1 DWORD per 128-DWORD row (=> LDS row stride of 129 floats,
// killing the 32-way bank conflict on p-indexed reads). Descriptor per
// CDNA5 ISA. Issued by one wave; tracked by TENSORcnt.
// ---------------------------------------------------------------------------
__device__ __forceinline__ void tdm_load_tile_f32(unsigned lds_addr,
                                                  const float* gptr,
                                                  unsigned rows) {
  unsigned long long ga = (unsigned long long)(size_t)gptr;
  u32x4 g0;
  g0[0] = 1u;                                   // [1:0] count=1, user mode
  g0[1] = lds_addr;                             // [63:32] lds_addr (bytes)
  g0[2] = (unsigned)(ga & 0xFFFFFFFFu);         // [95:64] global_addr lo
  g0[3] = (unsigned)((ga >> 32) & 0x1FFFFFFu)   // [120:96] global_addr hi
        | (2u << 30);                           // [127:126] type=2 ("image")
  i32x8 g1 = {};
  g1[0] = (2 << 16)        // data_size = 4 bytes
        | (1 << 20)        // pad_enable
        | (6 << 22)        // pad_interval: 2^6 8-byte units = 128 DWORDs
        | (0 << 25);       // pad_amount: 1 DWORD
  g1[1] = (int)(128u << 16);   // tensor_dim0 = 128   (bits [79:48])
  g1[2] = (int)(rows << 16);   // tensor_dim1 = rows  (bits [111:80])
  g1[3] = (int)(128u << 16);   // tile_dim0 = 128     (bits [127:112])
  g1[4] = (int)rows;           // tile_dim1 = rows    (bits [143:128])
  g1[5] = 128;                 // tensor_dim0_stride = 128 (bits [207:160])
  g1[6] = 0;                   // stride hi + tensor_dim1_stride (unused, 2D)
  g1[7] = 0;
  i32x4 z4 = {};
#if __clang_major__ >= 23
  i32x8 z8 = {};
  __builtin_amdgcn_tensor_load_to_lds(g0, g1, z4, z4, z8, 0);
#else
  __builtin_amdgcn_tensor_load_to_lds(g0, g1, z4, z4, 0);
#endif
}

// ---------------------------------------------------------------------------
// Kernel 4: fused bigram — never materialize proj [B,N,N,P] (268 MB saved).
// Block handles (b, 16 i's, 64 j's). Tiles are DMA'd into LDS by the Tensor
// Data Mover (wave 0 issues, waits TENSORcnt, then workgroup barrier). The
// TDM pad feature produces the 129-float LDS row stride directly.
// ---------------------------------------------------------------------------
__global__ void __launch_bounds__(256)
bigram_kernel(const float* __restrict__ left, const float* __restrict__ right,
              const float* __restrict__ Wo, const float* __restrict__ bo,
              float* __restrict__ out) {
  __shared__ float Lsh[64][PP + 1];
  __shared__ float Rsh[16][PP + 1];
  __shared__ float wos[PP];

  const int t = threadIdx.x;
  const int b     = blockIdx.z;   // 0..7
  const int itile = blockIdx.y;   // 0..15  -> 16 i's
  const int jtile = blockIdx.x;   // 0..3   -> 64 j's

  if (t < 32) {   // uniform wave-level branch: only wave 0 issues the DMAs
    tdm_load_tile_f32((unsigned)(size_t)&Lsh[0][0],
                      left + (size_t)(b * NN + jtile * 64) * PP, 64u);
    tdm_load_tile_f32((unsigned)(size_t)&Rsh[0][0],
                      right + (size_t)(b * NN + itile * 16) * PP, 16u);
    __builtin_amdgcn_s_wait_tensorcnt(0);
  }
  if (t < PP) wos[t] = Wo[t];
  __syncthreads();   // releases waves 1..7 only after wave 0 drained TENSORcnt

  const int jl = t & 63;
  const int ig = t >> 6;          // 0..3, covers i_local = ig*4 .. ig*4+3
  float acc[4] = {0.f, 0.f, 0.f, 0.f};

  for (int p = 0; p < PP; ++p) {
    float l = Lsh[jl][p];
    float w = wos[p];
#pragma unroll
    for (int ii = 0; ii < 4; ++ii) {
      acc[ii] = fmaf(w, fmaxf(l + Rsh[(ig << 2) + ii][p], 0.0f), acc[ii]);
    }
  }

  const float bov = bo[0];
  const int j = jtile * 64 + jl;
#pragma unroll
  for (int ii = 0; ii < 4; ++ii) {
    int i = itile * 16 + (ig << 2) + ii;
    out[(long)(b * NN + i) * NN + j] = acc[ii] + bov;
  }
}

// ---------------------------------------------------------------------------
extern "C" void kernel_launch(void* const* d_in, const int* in_sizes, int n_in,
                              void* d_out, int out_size, void* d_ws, size_t ws_size,
                              hipStream_t stream) {
  const int*   sent = (const int*)d_in[0];
  const float* emb  = (const float*)d_in[1];
  const float* Wl   = (const float*)d_in[2];
  const float* bl   = (const float*)d_in[3];
  const float* Wr   = (const float*)d_in[4];
  const float* Wo   = (const float*)d_in[5];
  const float* bo   = (const float*)d_in[6];
  const float* Ws1  = (const float*)d_in[7];
  const float* bs1  = (const float*)d_in[8];
  const float* Ws2  = (const float*)d_in[9];
  const float* bs2  = (const float*)d_in[10];
  const float* We1  = (const float*)d_in[11];
  const float* be1  = (const float*)d_in[12];
  const float* We2  = (const float*)d_in[13];
  const float* be2  = (const float*)d_in[14];

  // Workspace layout (floats): x | left | right | hs1 | he1  (~6.7 MB)
  float* ws    = (float*)d_ws;
  float* x     = ws;                       // 2048*300
  float* left  = x    + ROWS * DD;         // 2048*128
  float* right = left + ROWS * PP;
  float* hs1   = right + ROWS * PP;
  float* he1   = hs1  + ROWS * PP;

  // Output: (bigram [8,256,256], start [8,256,1], end [8,256,1]) concat
  float* bigram = (float*)d_out;
  float* start_o = bigram + BB * NN * NN;
  float* end_o   = start_o + BB * NN;

  gather_embed_kernel<<<(ROWS * DD + 255) / 256, 256, 0, stream>>>(sent, emb, x);

  fused_gemm_wmma<<<dim3(32, ROWS / 16), 32, 0, stream>>>(
      x, Wl, Wr, Ws1, We1, bl, bs1, be1, left, right, hs1, he1);

  startend_kernel<<<ROWS / 8, 256, 0, stream>>>(
      hs1, he1, Ws2, We2, bs2, be2, start_o, end_o);

  bigram_kernel<<<dim3(NN / 64, NN / 16, BB), 256, 0, stream>>>(
      left, right, Wo, bo, bigram);
}